// GATDenoiser_22196390986596
// MI455X (gfx1250) — compile-verified
//
#include <hip/hip_runtime.h>
#include <hip/hip_bf16.h>
#include <cstddef>

typedef __attribute__((ext_vector_type(2))) float v2f;
typedef __attribute__((ext_vector_type(8))) float v8f;

#define BB 8
#define NN 2048
#define BN (BB * NN)
#define KNB 33   // 32 neighbors + self
#define EPS 1e-5f

// ---------------------------------------------------------------------------
// sq[i] = sum_k x[i][k]^2
__global__ void sqnorm_kernel(const float* __restrict__ X, float* __restrict__ sq,
                              int F, int total) {
    int i = blockIdx.x * blockDim.x + threadIdx.x;
    if (i >= total) return;
    const float* x = X + (size_t)i * F;
    float a = 0.f;
    for (int k = 0; k < F; ++k) a += x[k] * x[k];
    sq[i] = a;
}

// ---------------------------------------------------------------------------
// One wave computes one 16x16 tile of D = sq_i + sq_j - 2 * X X^T via
// V_WMMA_F32_16X16X4_F32.  F is a compile-time constant so the K-loop has no
// bounds checks: each lane feeds the WMMA from one float2 (global_load_b64)
// per operand per step.  8 waves / block -> 8 tiles per block.
template <int F>
__global__ void gram_wmma_kernel(const float* __restrict__ X,
                                 const float* __restrict__ sq,
                                 float* __restrict__ D, int N) {
    int lane = threadIdx.x & 31;
    int wave = threadIdx.x >> 5;
    int tile = blockIdx.x * 8 + wave;
    int ntj  = N >> 4;
    int ti = tile / ntj, tj = tile % ntj;
    int i0 = ti << 4, j0 = tj << 4;
    int half = lane >> 4;      // 0: K pair {k,k+1}, 1: K pair {k+2,k+3}
    int m    = lane & 15;      // row (A) / col (B) within tile

    v8f c = {};
    const float* arow = X + (size_t)(i0 + m) * F;
    const float* brow = X + (size_t)(j0 + m) * F;

    if constexpr (F == 3) {
        // one K-step, 4th column zero-padded; guard resolves at compile time
        v2f a, b;
        a.x = arow[half << 1];
        a.y = half ? 0.f : arow[1];
        b.x = brow[half << 1];
        b.y = half ? 0.f : brow[1];
        c = __builtin_amdgcn_wmma_f32_16x16x4_f32(false, a, false, b,
                                                  (short)0, c, false, false);
    } else {
        const v2f* a2 = (const v2f*)arow;   // 8B-aligned: F multiple of 4
        const v2f* b2 = (const v2f*)brow;
#pragma unroll 4
        for (int kk = 0; kk < F / 4; ++kk) {
            v2f a = a2[(kk << 1) + half];
            v2f b = b2[(kk << 1) + half];
            c = __builtin_amdgcn_wmma_f32_16x16x4_f32(false, a, false, b,
                                                      (short)0, c, false, false);
        }
    }

    int col = j0 + m;          // C/D layout: lane 0-15 -> N=lane, 16-31 -> N=lane-16
    float sj = sq[col];
#pragma unroll
    for (int r = 0; r < 8; ++r) {
        int row = i0 + r + (half << 3);   // VGPR r: M=r (lanes 0-15), M=r+8 (16-31)
        float d = sq[row] + sj - 2.0f * c[r];
        if (row == col) d = 1e30f;        // mask diagonal (reference eye -> 1e30)
        D[(size_t)row * N + col] = d;
    }
}

// ---------------------------------------------------------------------------
// Per-row top-32 smallest distances (ascending), then append self index.
// Row cached in LDS; 32 x parallel argmin with lowest-index tie-break.
__global__ void topk_kernel(const float* __restrict__ D, int* __restrict__ idxOut,
                            int N) {
    __shared__ float vals[NN];
    __shared__ float rmin[256];
    __shared__ int   ridx[256];
    int row = blockIdx.x;
    int tid = threadIdx.x;
    const float* drow = D + (size_t)row * N;
    for (int j = tid; j < N; j += 256) vals[j] = drow[j];
    __syncthreads();
    int* out = idxOut + (size_t)row * KNB;
    for (int k = 0; k < 32; ++k) {
        float best = 3.0e38f; int bi = N;
        for (int j = tid; j < N; j += 256) {
            float v = vals[j];
            if (v < best || (v == best && j < bi)) { best = v; bi = j; }
        }
        rmin[tid] = best; ridx[tid] = bi;
        __syncthreads();
        for (int s = 128; s > 0; s >>= 1) {
            if (tid < s) {
                float ov = rmin[tid + s]; int oi = ridx[tid + s];
                if (ov < rmin[tid] || (ov == rmin[tid] && oi < ridx[tid])) {
                    rmin[tid] = ov; ridx[tid] = oi;
                }
            }
            __syncthreads();
        }
        if (tid == 0) { out[k] = ridx[0]; vals[ridx[0]] = 3.0e38f; }
        __syncthreads();
    }
    if (tid == 0) out[32] = row;   // self index (per-batch local)
}

// ---------------------------------------------------------------------------
// Y[r][o] = sum_k X[r][k] * W[k][o]
__global__ void matmul_kernel(const float* __restrict__ X, const float* __restrict__ W,
                              float* __restrict__ Y, int Fin, int Fout, int rows) {
    int i = blockIdx.x * blockDim.x + threadIdx.x;
    if (i >= rows * Fout) return;
    int r = i / Fout, o = i % Fout;
    const float* x = X + (size_t)r * Fin;
    float acc = 0.f;
    for (int k = 0; k < Fin; ++k) acc += x[k] * W[k * Fout + o];
    Y[i] = acc;
}

// ---------------------------------------------------------------------------
// s_dst/s_src per (row, head): dot of xw[row,h,:] with att[h][0/1][:]
__global__ void scores_kernel(const float* __restrict__ XW, const float* __restrict__ att,
                              float* __restrict__ sdst, float* __restrict__ ssrc,
                              int H, int C, int rows) {
    int i = blockIdx.x * blockDim.x + threadIdx.x;
    if (i >= rows * H) return;
    int r = i / H, h = i % H;
    const float* xw = XW + ((size_t)r * H + h) * C;
    const float* ad = att + (size_t)(h * 2 + 0) * C;
    const float* as = att + (size_t)(h * 2 + 1) * C;
    float d = 0.f, s = 0.f;
    for (int c = 0; c < C; ++c) { d += xw[c] * ad[c]; s += xw[c] * as[c]; }
    sdst[i] = d; ssrc[i] = s;
}

// ---------------------------------------------------------------------------
// GAT aggregation: softmax over 33 neighbors per head, weighted sum of xw_j.
// concat=1: out = prelu(concat_heads + bias).  concat=0: out = mean_heads + bias.
__global__ void gat_kernel(const float* __restrict__ XW, const int* __restrict__ idx,
                           const float* __restrict__ sdst, const float* __restrict__ ssrc,
                           const float* __restrict__ bias, const float* __restrict__ p,
                           float slope, int H, int C, int concat,
                           float* __restrict__ out, int N) {
    __shared__ float tmp[128];
    int bn = blockIdx.x;                 // global row in [0, B*N)
    int b  = bn / N;
    int t  = threadIdx.x;
    int HC = H * C;
    float res = 0.f;
    if (t < HC) {
        int h = t / C, c = t % C;
        const int* nb = idx + (size_t)bn * KNB;
        float sd = sdst[(size_t)bn * H + h];
        float m = -3.0e38f;
        for (int k = 0; k < KNB; ++k) {
            int j = b * N + nb[k];
            float e = sd + ssrc[(size_t)j * H + h];
            e = (e >= 0.f) ? e : slope * e;
            m = (e > m) ? e : m;
        }
        float num = 0.f, den = 0.f;
        for (int k = 0; k < KNB; ++k) {
            int j = b * N + nb[k];
            float e = sd + ssrc[(size_t)j * H + h];
            e = (e >= 0.f) ? e : slope * e;
            float w = __expf(e - m);
            den += w;
            num += w * XW[((size_t)j * H + h) * C + c];
        }
        res = num / den;
    }
    if (concat) {
        if (t < HC) {
            float v = res + bias[t];
            float pp = p[0];
            v = (v >= 0.f) ? v : pp * v;   // fused PReLU
            out[(size_t)bn * HC + t] = v;
        }
    } else {
        if (t < HC) tmp[t] = res;
        __syncthreads();
        if (t < C) {
            float a = 0.f;
            for (int h = 0; h < H; ++h) a += tmp[h * C + t];
            out[(size_t)bn * C + t] = a / (float)H + bias[t];
        }
    }
}

// ---------------------------------------------------------------------------
// Per-channel mean/var over (B,N). stats[c]=mean, stats[128+c]=var (biased).
__global__ void bn_stats_kernel(const float* __restrict__ Y, float* __restrict__ stats,
                                int Cf, int rows) {
    __shared__ float s1[256], s2[256];
    int c = blockIdx.x, tid = threadIdx.x;
    float a = 0.f, b = 0.f;
    for (int r = tid; r < rows; r += 256) {
        float v = Y[(size_t)r * Cf + c];
        a += v; b += v * v;
    }
    s1[tid] = a; s2[tid] = b;
    __syncthreads();
    for (int s = 128; s > 0; s >>= 1) {
        if (tid < s) { s1[tid] += s1[tid + s]; s2[tid] += s2[tid + s]; }
        __syncthreads();
    }
    if (tid == 0) {
        float mean = s1[0] / (float)rows;
        float var  = s2[0] / (float)rows - mean * mean;
        stats[c] = mean; stats[128 + c] = var;
    }
}

__global__ void bn_apply_kernel(float* __restrict__ Y, const float* __restrict__ stats,
                                const float* __restrict__ g, const float* __restrict__ be,
                                int Cf, int rows) {
    int i = blockIdx.x * blockDim.x + threadIdx.x;
    if (i >= rows * Cf) return;
    int c = i % Cf;
    float mean = stats[c], var = stats[128 + c];
    Y[i] = (Y[i] - mean) * rsqrtf(var + EPS) * g[c] + be[c];
}

// ---------------------------------------------------------------------------
__global__ void final_kernel(const float* __restrict__ H3, const float* __restrict__ Wc,
                             const float* __restrict__ bc, float* __restrict__ out,
                             int rows) {
    int i = blockIdx.x * blockDim.x + threadIdx.x;
    if (i >= rows * 16) return;
    int r = i / 16, o = i % 16;
    const float* h = H3 + (size_t)r * 6;
    float acc = bc[o];
    for (int k = 0; k < 6; ++k) acc += h[k] * Wc[k * 16 + o];
    out[i] = acc;
}

// ---------------------------------------------------------------------------
extern "C" void kernel_launch(void* const* d_in, const int* in_sizes, int n_in,
                              void* d_out, int out_size, void* d_ws, size_t ws_size,
                              hipStream_t stream) {
    (void)in_sizes; (void)n_in; (void)out_size; (void)ws_size;
    const float* x    = (const float*)d_in[0];
    // d_in[1] = target (unused by forward pass)
    const float* W1   = (const float*)d_in[2];
    const float* att1 = (const float*)d_in[3];
    const float* b1   = (const float*)d_in[4];
    const float* p1   = (const float*)d_in[5];
    const float* g1   = (const float*)d_in[6];
    const float* be1  = (const float*)d_in[7];
    const float* W2   = (const float*)d_in[8];
    const float* att2 = (const float*)d_in[9];
    const float* b2   = (const float*)d_in[10];
    const float* p2   = (const float*)d_in[11];
    const float* g2   = (const float*)d_in[12];
    const float* be2  = (const float*)d_in[13];
    const float* W3   = (const float*)d_in[14];
    const float* att3 = (const float*)d_in[15];
    const float* b3   = (const float*)d_in[16];
    const float* Wc   = (const float*)d_in[17];
    const float* bc   = (const float*)d_in[18];

    char* ws = (char*)d_ws;
    float* D    = (float*)(ws + 0);          // 16 MB: one batch's NxN distance matrix
    int*   idx  = (int*)  (ws + 16777216);   // B*N*33 ints
    float* sq   = (float*)(ws + 18939904);   // B*N
    float* xw   = (float*)(ws + 19005440);   // B*N*128 max
    float* sdst = (float*)(ws + 27394048);   // B*N*8
    float* ssrc = (float*)(ws + 27918336);   // B*N*8
    float* hA   = (float*)(ws + 28442624);   // B*N*128 ping
    float* hB   = (float*)(ws + 36831232);   // B*N*128 pong
    float* st   = (float*)(ws + 45219840);   // 256 floats of BN stats

    const int gramGrid = (NN / 16) * (NN / 16) / 8;

    // ---- Layer 1: F 3 -> 64, H=4, C=16, concat, PReLU+BN ----
    sqnorm_kernel<<<(BN + 255) / 256, 256, 0, stream>>>(x, sq, 3, BN);
    for (int b = 0; b < BB; ++b) {
        gram_wmma_kernel<3><<<gramGrid, 256, 0, stream>>>(
            x + (size_t)b * NN * 3, sq + (size_t)b * NN, D, NN);
        topk_kernel<<<NN, 256, 0, stream>>>(D, idx + (size_t)b * NN * KNB, NN);
    }
    matmul_kernel<<<(BN * 64 + 255) / 256, 256, 0, stream>>>(x, W1, xw, 3, 64, BN);
    scores_kernel<<<(BN * 4 + 255) / 256, 256, 0, stream>>>(xw, att1, sdst, ssrc, 4, 16, BN);
    gat_kernel<<<BN, 64, 0, stream>>>(xw, idx, sdst, ssrc, b1, p1, 0.2f, 4, 16, 1, hA, NN);
    bn_stats_kernel<<<64, 256, 0, stream>>>(hA, st, 64, BN);
    bn_apply_kernel<<<(BN * 64 + 255) / 256, 256, 0, stream>>>(hA, st, g1, be1, 64, BN);

    // ---- Layer 2: F 64 -> 128, H=2, C=64, concat, PReLU+BN ----
    sqnorm_kernel<<<(BN + 255) / 256, 256, 0, stream>>>(hA, sq, 64, BN);
    for (int b = 0; b < BB; ++b) {
        gram_wmma_kernel<64><<<gramGrid, 256, 0, stream>>>(
            hA + (size_t)b * NN * 64, sq + (size_t)b * NN, D, NN);
        topk_kernel<<<NN, 256, 0, stream>>>(D, idx + (size_t)b * NN * KNB, NN);
    }
    matmul_kernel<<<(BN * 128 + 255) / 256, 256, 0, stream>>>(hA, W2, xw, 64, 128, BN);
    scores_kernel<<<(BN * 2 + 255) / 256, 256, 0, stream>>>(xw, att2, sdst, ssrc, 2, 64, BN);
    gat_kernel<<<BN, 128, 0, stream>>>(xw, idx, sdst, ssrc, b2, p2, 0.2f, 2, 64, 1, hB, NN);
    bn_stats_kernel<<<128, 256, 0, stream>>>(hB, st, 128, BN);
    bn_apply_kernel<<<(BN * 128 + 255) / 256, 256, 0, stream>>>(hB, st, g2, be2, 128, BN);

    // ---- Layer 3: F 128 -> 48, H=8, C=6, mean over heads ----
    sqnorm_kernel<<<(BN + 255) / 256, 256, 0, stream>>>(hB, sq, 128, BN);
    for (int b = 0; b < BB; ++b) {
        gram_wmma_kernel<128><<<gramGrid, 256, 0, stream>>>(
            hB + (size_t)b * NN * 128, sq + (size_t)b * NN, D, NN);
        topk_kernel<<<NN, 256, 0, stream>>>(D, idx + (size_t)b * NN * KNB, NN);
    }
    matmul_kernel<<<(BN * 48 + 255) / 256, 256, 0, stream>>>(hB, W3, xw, 128, 48, BN);
    scores_kernel<<<(BN * 8 + 255) / 256, 256, 0, stream>>>(xw, att3, sdst, ssrc, 8, 6, BN);
    gat_kernel<<<BN, 64, 0, stream>>>(xw, idx, sdst, ssrc, b3, nullptr, 0.5f, 8, 6, 0, hA, NN);

    // ---- Classifier: 6 -> 16 ----
    final_kernel<<<(BN * 16 + 255) / 256, 256, 0, stream>>>(hA, Wc, bc, (float*)d_out, BN);
}